// attribute_net_30382598652298
// MI455X (gfx1250) — compile-verified
//
#include <hip/hip_runtime.h>
#include <hip/hip_bf16.h>

typedef float v2f __attribute__((ext_vector_type(2)));
typedef float v8f __attribute__((ext_vector_type(8)));

#define B_SZ    64
#define R_ROWS  196
#define MFEAT   2048
#define ACLS    320
#define TCLS    1000

#define KC      32          // K chunk staged in LDS
#define KCP     36          // padded LDS row stride (words): bank-conflict-free, 16B aligned
#define NWAVES  20          // one wave per 16-wide N tile (20*16 = 320)
#define BLK2    (NWAVES*32) // 640 threads
#define MT      4           // 16-row tiles per block (M block = 64 rows)
#define MBLK    (MT*16)
#define MBLOCKS 4           // ceil(196/64)

__device__ inline float waveMax(float v) {
#pragma unroll
    for (int off = 16; off >= 1; off >>= 1)
        v = fmaxf(v, __shfl_xor(v, off, 32));
    return v;
}
__device__ inline float waveSum(float v) {
#pragma unroll
    for (int off = 16; off >= 1; off >>= 1)
        v += __shfl_xor(v, off, 32);
    return v;
}

__global__ void zero_kernel(float* __restrict__ p, int n) {
    int i = blockIdx.x * blockDim.x + threadIdx.x;
    if (i < n) p[i] = 0.0f;
}

// Branch-free K loop, specialized on the number of active 16-row tiles (MTN = 4 or 1).
// Selecting the instantiation once per block (uniform) avoids per-WMMA accumulator
// phi-merges / v_mov storms while still skipping dead tiles in the last block.
template <int MTN>
__device__ __attribute__((always_inline))
void kloop(const float* __restrict__ Abase, const float* __restrict__ W1,
           float* As, float* Bs, v8f acc[MT],
           int m0, int t, int lr, int lh, int n0) {
    for (int k0 = 0; k0 < MFEAT; k0 += KC) {
        __syncthreads();
        // Stage A tile: MBLK x KC f32 as float4 (rows >=196 zero-padded)
        if (t < (MBLK * KC) / 4) {          // 512 threads
            int row  = t >> 3;
            int col4 = (t & 7) << 2;
            float4 v = make_float4(0.f, 0.f, 0.f, 0.f);
            int gr = m0 + row;
            if (gr < R_ROWS)
                v = *(const float4*)(Abase + (size_t)gr * MFEAT + k0 + col4);
            *(float4*)(&As[row * KCP + col4]) = v;
        }
        // Stage W1 chunk: 320 x KC f32 as float4 (W1 row-major [a][m])
        for (int i = t; i < (ACLS * KC) / 4; i += BLK2) {
            int row  = i >> 3;
            int col4 = (i & 7) << 2;
            float4 v = *(const float4*)(W1 + (size_t)row * MFEAT + k0 + col4);
            *(float4*)(&Bs[row * KCP + col4]) = v;
        }
        __syncthreads();
#pragma unroll
        for (int ks = 0; ks < KC; ks += 4) {
            const int kk = ks + (lh << 1);
            v2f bb;
            bb.x = Bs[(n0 + lr) * KCP + kk];
            bb.y = Bs[(n0 + lr) * KCP + kk + 1];
#pragma unroll
            for (int mt = 0; mt < MTN; ++mt) {
                v2f a;
                a.x = As[(mt * 16 + lr) * KCP + kk];
                a.y = As[(mt * 16 + lr) * KCP + kk + 1];
                acc[mt] = __builtin_amdgcn_wmma_f32_16x16x4_f32(
                    false, a, false, bb, (short)0, acc[mt], false, false);
            }
        }
    }
}

// One block per (batch b, 64-row tile of R). 20 waves; wave w owns N columns [16w,16w+16)
// across MT=4 stacked 16x16 output tiles (amortizes W1/LDS B-fragment reuse 4x).
// logits = A(64x2048) @ W1^T(2048x320) via v_wmma_f32_16x16x4_f32, then per-row softmax
// over 320 (tile-by-tile through LDS), masked row-sum -> atomicAdd attr_dis[b][*]/320.
__global__ __launch_bounds__(BLK2)
void gemm1_softmax_sum(const float* __restrict__ attr_map,
                       const float* __restrict__ W1,
                       float* __restrict__ attr_dis) {
    __shared__ float As[MBLK * KCP];        // 9216 B
    __shared__ float Bs[ACLS * KCP];        // 46080 B ; aliased as Lg[16][320] afterwards
    __shared__ float colsum[ACLS];          // 1280 B

    const int b     = blockIdx.y;           // 0..63
    const int m0    = blockIdx.x * MBLK;    // 0,64,128,192
    const int t     = threadIdx.x;
    const int wave  = t >> 5;
    const int lane  = t & 31;
    const int lr    = lane & 15;            // M (A) / N (B) index within tile
    const int lh    = lane >> 4;            // selects K pair {0,1} vs {2,3}
    const int n0    = wave * 16;
    const int mtv   = min(MT, (R_ROWS - m0 + 15) >> 4);   // 4 for blocks 0-2, 1 for block 3

    v8f acc[MT];
#pragma unroll
    for (int mt = 0; mt < MT; ++mt) acc[mt] = (v8f){};

    const float* Abase = attr_map + (size_t)b * R_ROWS * MFEAT;

    if (mtv == MT)
        kloop<MT>(Abase, W1, As, Bs, acc, m0, t, lr, lh, n0);
    else
        kloop<1>(Abase, W1, As, Bs, acc, m0, t, lr, lh, n0);

    // ---- softmax + region sum, one 16-row tile at a time through Lg (aliases Bs) ----
    float* Lg = Bs;                          // 16*320 floats needed <= Bs capacity
    for (int i = t; i < ACLS; i += BLK2) colsum[i] = 0.0f;
#pragma unroll
    for (int mt = 0; mt < MT; ++mt) {
        __syncthreads();                     // Bs/Lg reads of previous phase complete
        const int mbase = lh << 3;           // C layout: M = v + 8*lh, N = n0+lr
#pragma unroll
        for (int v = 0; v < 8; ++v)
            Lg[(mbase + v) * ACLS + n0 + lr] = acc[mt][v];
        __syncthreads();
        const int grow = m0 + mt * 16 + wave;
        if (wave < 16 && grow < R_ROWS) {
            const float* row = &Lg[wave * ACLS];
            float mx = -3.4e38f;
            for (int i = lane; i < ACLS; i += 32) mx = fmaxf(mx, row[i]);
            mx = waveMax(mx);
            float s = 0.0f;
            for (int i = lane; i < ACLS; i += 32) s += expf(row[i] - mx);
            s = waveSum(s);
            const float inv = 1.0f / s;
            for (int i = lane; i < ACLS; i += 32)
                atomicAdd(&colsum[i], expf(row[i] - mx) * inv);
        }
    }
    __syncthreads();
    for (int i = t; i < ACLS; i += BLK2)
        atomicAdd(&attr_dis[(size_t)b * ACLS + i], colsum[i] * (1.0f / (float)ACLS));
}

// One block per batch row: attr_final = softmax(attr_dis @ W2^T),
// map_final = softmax(features @ W3^T), final = 0.5*(attr_final + map_final).
__global__ __launch_bounds__(1024)
void head_kernel(const float* __restrict__ features,
                 const float* __restrict__ W2,
                 const float* __restrict__ W3,
                 const float* __restrict__ attr_dis,
                 float* __restrict__ final_out) {
    __shared__ float feat[MFEAT];   // 8 KB
    __shared__ float adis[ACLS];    // 1.25 KB
    __shared__ float buf[TCLS];     // logits / exp scratch
    __shared__ float sred;

    const int b    = blockIdx.x;
    const int t    = threadIdx.x;
    const int lane = t & 31;

    for (int i = t; i < MFEAT; i += 1024) feat[i] = features[(size_t)b * MFEAT + i];
    for (int i = t; i < ACLS;  i += 1024) adis[i] = attr_dis[(size_t)b * ACLS + i];
    __syncthreads();

    // ---- path 1: attr_dis @ W2^T -> softmax ----
    if (t < TCLS) {
        const float* w = W2 + (size_t)t * ACLS;
        float a = 0.0f;
        for (int k = 0; k < ACLS; ++k) a = fmaf(adis[k], w[k], a);
        buf[t] = a;
    }
    __syncthreads();
    if (t < 32) {
        float mx = -3.4e38f;
        for (int i = lane; i < TCLS; i += 32) mx = fmaxf(mx, buf[i]);
        mx = waveMax(mx);
        if (lane == 0) sred = mx;
    }
    __syncthreads();
    float e1 = 0.0f;
    {
        const float mx = sred;
        if (t < TCLS) { e1 = expf(buf[t] - mx); buf[t] = e1; }
    }
    __syncthreads();
    if (t < 32) {
        float s = 0.0f;
        for (int i = lane; i < TCLS; i += 32) s += buf[i];
        s = waveSum(s);
        if (lane == 0) sred = s;
    }
    __syncthreads();
    const float afinal = e1 / sred;
    __syncthreads();

    // ---- path 2: features @ W3^T -> softmax ----
    if (t < TCLS) {
        const float* w = W3 + (size_t)t * MFEAT;
        float a = 0.0f;
        for (int k = 0; k < MFEAT; ++k) a = fmaf(feat[k], w[k], a);
        buf[t] = a;
    }
    __syncthreads();
    if (t < 32) {
        float mx = -3.4e38f;
        for (int i = lane; i < TCLS; i += 32) mx = fmaxf(mx, buf[i]);
        mx = waveMax(mx);
        if (lane == 0) sred = mx;
    }
    __syncthreads();
    float e2 = 0.0f;
    {
        const float mx = sred;
        if (t < TCLS) { e2 = expf(buf[t] - mx); buf[t] = e2; }
    }
    __syncthreads();
    if (t < 32) {
        float s = 0.0f;
        for (int i = lane; i < TCLS; i += 32) s += buf[i];
        s = waveSum(s);
        if (lane == 0) sred = s;
    }
    __syncthreads();
    if (t < TCLS) {
        const float mfinal = e2 / sred;
        final_out[(size_t)b * TCLS + t] = 0.5f * (afinal + mfinal);
    }
}

extern "C" void kernel_launch(void* const* d_in, const int* in_sizes, int n_in,
                              void* d_out, int out_size, void* d_ws, size_t ws_size,
                              hipStream_t stream) {
    (void)in_sizes; (void)n_in; (void)out_size; (void)d_ws; (void)ws_size;
    const float* attr_map = (const float*)d_in[0];  // [64,196,2048]
    const float* features = (const float*)d_in[1];  // [64,2048]
    const float* W1       = (const float*)d_in[2];  // [320,2048]
    const float* W2       = (const float*)d_in[3];  // [1000,320]
    const float* W3       = (const float*)d_in[4];  // [1000,2048]

    float* out       = (float*)d_out;
    float* final_out = out;                       // [64,1000]
    float* attr_dis  = out + B_SZ * TCLS;         // [64,320] (also 2nd tuple output)

    const int nAttr = B_SZ * ACLS;
    zero_kernel<<<(nAttr + 255) / 256, 256, 0, stream>>>(attr_dis, nAttr);

    dim3 g2(MBLOCKS, B_SZ);
    gemm1_softmax_sum<<<g2, BLK2, 0, stream>>>(attr_map, W1, attr_dis);

    head_kernel<<<B_SZ, 1024, 0, stream>>>(features, W2, W3, attr_dis, final_out);
}